// CrossLayer_34686155882557
// MI455X (gfx1250) — compile-verified
//
#include <hip/hip_runtime.h>
#include <hip/hip_bf16.h>

// CDNA5 / gfx1250 wave32.
typedef __attribute__((ext_vector_type(2))) float v2f;
typedef __attribute__((ext_vector_type(4))) float v4f;
typedef __attribute__((ext_vector_type(8))) float v8f;

#define D_DIM 1024
#define ROWS_PER_BLOCK 16   // one 16-row WMMA tile per block
#define WAVES_PER_BLOCK 8   // each wave reduces a 128-wide K slice

// out[b,d] = x0[b,d] * (xl[b,:]·w) + bias[d] + xl[b,d]
//
// Phase 1 (per wave): V_WMMA_F32_16X16X4_F32 with
//   A[m,k] = w[k0+k]        (broadcast across M; exact f32)
//   B[k,n] = xl[row n][k0+k]
// => D[m,n] = partial s_n. Lane L ends with c[0] = partial dot of row (L&15).
// Phase 2: LDS-reduce the 8 wave partials, then float4 elementwise epilogue.
__global__ __launch_bounds__(256) void crosslayer_kernel(
    const float* __restrict__ x0, const float* __restrict__ xl,
    const float* __restrict__ w,  const float* __restrict__ bias,
    float* __restrict__ out)
{
    __shared__ float part[WAVES_PER_BLOCK][ROWS_PER_BLOCK];
    __shared__ float srow[ROWS_PER_BLOCK];

    const int tid  = threadIdx.x;
    const int wave = tid >> 5;          // 0..7  (wave32)
    const int lane = tid & 31;
    const int half = lane >> 4;         // 0: K=0,1  1: K=2,3 (per WMMA f32 layout)
    const int r    = lane & 15;         // row within 16-row tile
    const size_t row0 = (size_t)blockIdx.x * ROWS_PER_BLOCK;

    // Warm the x0 tile for the epilogue while the dot phase runs.
    // One prefetch per 128B line: threads stride col by 16B, so pick 1 of 8.
    if ((tid & 7) == 0) {
        const size_t col = (size_t)tid * 4;
#pragma unroll
        for (int i = 0; i < ROWS_PER_BLOCK; ++i)
            __builtin_prefetch(x0 + (row0 + i) * D_DIM + col, 0, 3);
    }

    // ---- Phase 1: K-sliced GEMV via f32 WMMA -------------------------------
    v8f c = {};  // D accumulator; c[0] lane L = partial s_{L&15}
    const float* xlrow = xl + (row0 + r) * (size_t)D_DIM;
    const int kbase = wave * (D_DIM / WAVES_PER_BLOCK) + 2 * half;

#pragma unroll 4
    for (int j = 0; j < (D_DIM / WAVES_PER_BLOCK) / 4; ++j) {
        const int k = kbase + 4 * j;
        v2f a = *(const v2f*)(w + k);       // A: w chunk, broadcast over M
        v2f b = *(const v2f*)(xlrow + k);   // B: xl rows, transposed layout
        // 8 args: (neg_a, A, neg_b, B, c_mod, C, reuse_a, reuse_b)
        c = __builtin_amdgcn_wmma_f32_16x16x4_f32(
                false, a, false, b, (short)0, c, false, false);
    }

    if (lane < 16) part[wave][lane] = c[0];
    __syncthreads();

    if (tid < ROWS_PER_BLOCK) {
        float s = 0.0f;
#pragma unroll
        for (int wv = 0; wv < WAVES_PER_BLOCK; ++wv) s += part[wv][tid];
        srow[tid] = s;
    }
    __syncthreads();

    // ---- Phase 2: elementwise epilogue, float4 streams ---------------------
    // 256 threads x float4 = 1024 floats = exactly one row per iteration.
    const size_t col = (size_t)tid * 4;
    const v4f bb = *(const v4f*)(bias + col);   // bias is [D,1] -> flat D
#pragma unroll 4
    for (int i = 0; i < ROWS_PER_BLOCK; ++i) {
        const size_t off = (row0 + i) * D_DIM + col;
        const float s = srow[i];
        v4f a = *(const v4f*)(x0 + off);
        v4f l = *(const v4f*)(xl + off);   // hits WGP$/L2 (just streamed)
        v4f o = a * s + bb + l;
        *(v4f*)(out + off) = o;
    }
}

extern "C" void kernel_launch(void* const* d_in, const int* in_sizes, int n_in,
                              void* d_out, int out_size, void* d_ws, size_t ws_size,
                              hipStream_t stream) {
    const float* x0   = (const float*)d_in[0];  // [B, D]
    const float* xl   = (const float*)d_in[1];  // [B, D]
    const float* kern = (const float*)d_in[2];  // [D, 1]
    const float* bias = (const float*)d_in[3];  // [D, 1]
    float* out = (float*)d_out;                 // [B, D]

    const int B = in_sizes[0] / D_DIM;          // 16384
    const int grid = B / ROWS_PER_BLOCK;        // 1024 blocks
    crosslayer_kernel<<<grid, 256, 0, stream>>>(x0, xl, kern, bias, out);
}